// BEMv13StatefulModule_57226144252187
// MI455X (gfx1250) — compile-verified
//
#include <hip/hip_runtime.h>

typedef float v2f __attribute__((ext_vector_type(2)));
typedef float v4f __attribute__((ext_vector_type(4)));
typedef float v8f __attribute__((ext_vector_type(8)));

#define M_TOT  16384   // B*S = 4*4096
#define D_DIM  2048
#define O_DIM  2048
#define E_NUM  2
#define R_RK   8
#define LORA_SCALE 2.0f   // 16/8

#define BM 128
#define BN 128
#define BK 32
#define LDSS 36          // LDS row stride in floats: 144B -> 16B aligned rows, conflict-free

__global__ __launch_bounds__(256)
void fused_lora_moe_kernel(const float* __restrict__ x,
                           const float* __restrict__ Wb,
                           const float* __restrict__ bb,
                           const float* __restrict__ Amat,   // [E,R,D] == [16, D]
                           const float* __restrict__ Bmat,   // [E,O,R]
                           const float* __restrict__ Wr,     // [D,E]
                           const float* __restrict__ br,     // [E]
                           float* __restrict__ out)
{
    __shared__ float xs[BM * LDSS];   // x tile      [128 x 32]
    __shared__ float ws[BN * LDSS];   // W_base tile [128 x 32]
    __shared__ float ss[32 * LDSS];   // small mat   [32 x 32] (A rows 0..15, router 16..17, zeros)
    __shared__ float gb[BM * LDSS];   // staged h + router logits per token

    const int tid   = threadIdx.x;
    const int lane  = tid & 31;
    const int wave  = tid >> 5;
    const int lrow  = lane & 15;        // row/col within a 16-wide fragment
    const int halfk = (lane >> 4) * 2;  // 0 for lanes 0-15, 2 for lanes 16-31
    const int mBase = blockIdx.y * BM;
    const int nBase = blockIdx.x * BN;

    // per-thread staging coordinates (constant across K-chunks)
    const int g_row = tid >> 3;         // 0..31 within a 32-row group
    const int g_c   = (tid & 7) * 4;    // 0,4,...,28

    v8f acc[8];
    v8f accS[2];
    const v8f zero8 = {0.f,0.f,0.f,0.f,0.f,0.f,0.f,0.f};
#pragma unroll
    for (int i = 0; i < 8; ++i) acc[i] = zero8;
    accS[0] = zero8; accS[1] = zero8;

    // register prefetch buffers (software pipeline, depth 2)
    v4f rx[4], rw[4], rs;

#define LOAD_GLOBALS(K0)                                                       \
    do {                                                                       \
        _Pragma("unroll")                                                      \
        for (int i = 0; i < 4; ++i) {                                          \
            int idx = tid + i * 256;                                           \
            int row = idx >> 3;                                                \
            int c   = (idx & 7) * 4;                                           \
            rx[i] = *(const v4f*)(x  + (size_t)(mBase + row) * D_DIM + (K0) + c); \
            rw[i] = *(const v4f*)(Wb + (size_t)(nBase + row) * D_DIM + (K0) + c); \
        }                                                                      \
        if (g_row < 16) {                                                      \
            rs = *(const v4f*)(Amat + (size_t)g_row * D_DIM + (K0) + g_c);     \
        } else if (g_row < 18) {                                               \
            int e = g_row - 16;                                                \
            rs.x = Wr[(size_t)((K0) + g_c + 0) * E_NUM + e];                   \
            rs.y = Wr[(size_t)((K0) + g_c + 1) * E_NUM + e];                   \
            rs.z = Wr[(size_t)((K0) + g_c + 2) * E_NUM + e];                   \
            rs.w = Wr[(size_t)((K0) + g_c + 3) * E_NUM + e];                   \
        } else {                                                               \
            rs.x = rs.y = rs.z = rs.w = 0.f;                                   \
        }                                                                      \
    } while (0)

    LOAD_GLOBALS(0);

    for (int k0 = 0; k0 < D_DIM; k0 += BK) {
        __syncthreads();   // previous chunk's LDS consumers done

        // ---- commit prefetched registers to LDS ----
#pragma unroll
        for (int i = 0; i < 4; ++i) {
            int idx = tid + i * 256;
            int row = idx >> 3;
            int c   = (idx & 7) * 4;
            *(v4f*)&xs[row * LDSS + c] = rx[i];
            *(v4f*)&ws[row * LDSS + c] = rw[i];
        }
        *(v4f*)&ss[g_row * LDSS + g_c] = rs;
        __syncthreads();

        // ---- prefetch next chunk's globals during compute ----
        if (k0 + BK < D_DIM) {
            LOAD_GLOBALS(k0 + BK);
        }

        // ---- compute: 8 K-steps of V_WMMA_F32_16X16X4_F32 per 16x16 tile ----
        const float* xrow = &xs[(wave * 16 + lrow) * LDSS];
#pragma unroll
        for (int kk = 0; kk < BK; kk += 4) {
            v2f a = *(const v2f*)&xrow[kk + halfk];
#pragma unroll
            for (int nt = 0; nt < 8; ++nt) {
                v2f b = *(const v2f*)&ws[(nt * 16 + lrow) * LDSS + kk + halfk];
                acc[nt] = __builtin_amdgcn_wmma_f32_16x16x4_f32(
                    false, a, false, b, (short)0, acc[nt], false, false);
            }
#pragma unroll
            for (int st = 0; st < 2; ++st) {
                v2f b = *(const v2f*)&ss[(st * 16 + lrow) * LDSS + kk + halfk];
                accS[st] = __builtin_amdgcn_wmma_f32_16x16x4_f32(
                    false, a, false, b, (short)0, accS[st], false, false);
            }
        }
    }
#undef LOAD_GLOBALS

    // ---- stage h (cols 0..15) and router logits (cols 16,17) into LDS, C/D layout ----
#pragma unroll
    for (int st = 0; st < 2; ++st) {
#pragma unroll
        for (int i = 0; i < 8; ++i) {
            int row = wave * 16 + (lane >> 4) * 8 + i;
            gb[row * LDSS + st * 16 + lrow] = accS[st][i];
        }
    }
    __syncthreads();

    // ---- per-token softmax over 2 experts, fold routing weight * scale into A-frag ----
    {
        int row = wave * 16 + lrow;
        float l0 = gb[row * LDSS + 16] + br[0];
        float l1 = gb[row * LDSS + 17] + br[1];
        float mx = fmaxf(l0, l1);
        float e0 = __expf(l0 - mx);
        float e1 = __expf(l1 - mx);
        float inv = 1.0f / (e0 + e1);
        float w0 = e0 * inv * LORA_SCALE;
        float w1 = e1 * inv * LORA_SCALE;

        // routed = G @ Bflat^T, K = E*R = 16 -> 4 WMMA K-steps, accumulated into acc[]
#pragma unroll
        for (int kk = 0; kk < 16; kk += 4) {
            int kidx = kk + halfk;               // even; pair {kidx, kidx+1} within one expert
            float wf = (kidx < R_RK) ? w0 : w1;
            v2f a;
            a.x = gb[row * LDSS + kidx]     * wf;
            a.y = gb[row * LDSS + kidx + 1] * wf;
            int e = kidx >> 3;
            int r = kidx & 7;
#pragma unroll
            for (int nt = 0; nt < 8; ++nt) {
                int o = nBase + nt * 16 + lrow;
                v2f b = *(const v2f*)(Bmat + ((size_t)e * O_DIM + o) * R_RK + r);
                acc[nt] = __builtin_amdgcn_wmma_f32_16x16x4_f32(
                    false, a, false, b, (short)0, acc[nt], false, false);
            }
        }
    }

    // ---- bias + non-temporal store (out is write-once; keep x/W resident in L2) ----
#pragma unroll
    for (int nt = 0; nt < 8; ++nt) {
        int col = nBase + nt * 16 + lrow;
        float bias = bb[col];
#pragma unroll
        for (int i = 0; i < 8; ++i) {
            int r = mBase + wave * 16 + (lane >> 4) * 8 + i;
            __builtin_nontemporal_store(acc[nt][i] + bias, &out[(size_t)r * O_DIM + col]);
        }
    }
}

extern "C" void kernel_launch(void* const* d_in, const int* in_sizes, int n_in,
                              void* d_out, int out_size, void* d_ws, size_t ws_size,
                              hipStream_t stream) {
    const float* x    = (const float*)d_in[0];  // [4,4096,2048]
    const float* Wb   = (const float*)d_in[1];  // [2048,2048]
    const float* bb   = (const float*)d_in[2];  // [2048]
    const float* Amat = (const float*)d_in[3];  // [2,8,2048]
    const float* Bmat = (const float*)d_in[4];  // [2,2048,8]
    const float* Wr   = (const float*)d_in[5];  // [2048,2]
    const float* br   = (const float*)d_in[6];  // [2]
    float* out = (float*)d_out;                 // [4,4096,2048]

    dim3 grid(O_DIM / BN, M_TOT / BM);          // (16, 128)
    dim3 block(256);
    fused_lora_moe_kernel<<<grid, block, 0, stream>>>(x, Wb, bb, Amat, Bmat, Wr, br, out);
}